// QLinear_60344290508883
// MI455X (gfx1250) — compile-verified
//
#include <hip/hip_runtime.h>
#include <hip/hip_bf16.h>

// fp32 GEMM + bias:  out[M,N] = X[M,K] * W[N,K]^T + b[N]
// (float_quantize with exp=8/man=23 is identity on fp32 -> plain fp32 matmul).
// CDNA5 path: V_WMMA_F32_16X16X4_F32 (wave32) + GLOBAL_LOAD_ASYNC_TO_LDS_B128
// staging tracked with ASYNCcnt (s_wait_asynccnt), double-buffered LDS.

typedef float v2f __attribute__((ext_vector_type(2)));
typedef float v8f __attribute__((ext_vector_type(8)));

static constexpr int M_DIM = 8192;
static constexpr int N_DIM = 4096;
static constexpr int K_DIM = 1024;

static constexpr int TM  = 128;      // block tile rows (M)
static constexpr int TN  = 128;      // block tile cols (N)
static constexpr int BK  = 8;        // K slab per stage
static constexpr int LDP = BK + 4;   // padded LDS stride (floats): conflict-free, 16B-aligned

__global__ __launch_bounds__(256)
void qlinear_wmma_f32(const float* __restrict__ X,
                      const float* __restrict__ W,
                      const float* __restrict__ bias,
                      float* __restrict__ out)
{
    // double-buffered operand tiles: 2 * (128 + 128) * 12 * 4B = 24 KB
    __shared__ float As[2][TM][LDP];
    __shared__ float Bs[2][TN][LDP];

    const int tid  = threadIdx.x;
    const int lane = tid & 31;
    const int wave = tid >> 5;

    const int m0 = blockIdx.y * TM;
    const int n0 = blockIdx.x * TN;

    // Global->LDS staging: each of 256 threads moves one b128 of A and B per slab.
    const int lr = tid >> 1;        // slab row 0..127
    const int lc = (tid & 1) * 4;   // 0 or 4 (float4 column)
    const float* Ap = X + (size_t)(m0 + lr) * K_DIM + lc;
    const float* Bp = W + (size_t)(n0 + lr) * K_DIM + lc;

    // LDS byte offsets for this thread's staging slot (generic ptr low 32 bits
    // == LDS offset per the gfx1250 LDS aperture mapping).
    const unsigned aoff0 = (unsigned)(size_t)&As[0][lr][lc];
    const unsigned boff0 = (unsigned)(size_t)&Bs[0][lr][lc];
    const unsigned abufs = (unsigned)(sizeof(float) * TM * LDP);  // As[1]-As[0]
    const unsigned bbufs = (unsigned)(sizeof(float) * TN * LDP);  // Bs[1]-Bs[0]

    // wave sub-tile: 2 waves across M (64 rows each), 4 waves across N (32 cols each)
    const int wm = (wave & 1) * 64;
    const int wn = (wave >> 1) * 32;
    const int la = lane & 15;       // row (A) / col (B) within 16-wide fragment
    const int kh = lane >> 4;       // K-pair selector: lanes 16-31 hold K=2,3

    const v8f vzero = {0.f, 0.f, 0.f, 0.f, 0.f, 0.f, 0.f, 0.f};
    v8f acc[4][2];
    #pragma unroll
    for (int i = 0; i < 4; ++i)
        #pragma unroll
        for (int j = 0; j < 2; ++j)
            acc[i][j] = vzero;

    // prologue: async-stage slab 0 straight into LDS buffer 0 (ASYNCcnt)
    asm volatile("global_load_async_to_lds_b128 %0, %1, off"
                 :: "v"(aoff0), "v"((unsigned long long)(size_t)Ap) : "memory");
    asm volatile("global_load_async_to_lds_b128 %0, %1, off"
                 :: "v"(boff0), "v"((unsigned long long)(size_t)Bp) : "memory");

    constexpr int NK = K_DIM / BK;  // 128 stages
    for (int kt = 0; kt < NK; ++kt) {
        const int cur = kt & 1;

        // our slab-kt async copies are complete; publish + retire old reads
        asm volatile("s_wait_asynccnt 0x0" ::: "memory");
        __syncthreads();

        // async-stage slab kt+1 into the other buffer (overlaps with WMMA below)
        if (kt + 1 < NK) {
            const unsigned nb = (unsigned)((kt + 1) & 1);
            const float* An = Ap + (kt + 1) * BK;
            const float* Bn = Bp + (kt + 1) * BK;
            asm volatile("global_load_async_to_lds_b128 %0, %1, off"
                         :: "v"(aoff0 + nb * abufs),
                            "v"((unsigned long long)(size_t)An) : "memory");
            asm volatile("global_load_async_to_lds_b128 %0, %1, off"
                         :: "v"(boff0 + nb * bbufs),
                            "v"((unsigned long long)(size_t)Bn) : "memory");
            if (kt + 2 < NK) {
                __builtin_prefetch(Ap + (kt + 2) * BK, 0, 3);   // global_prefetch_b8 (near)
                __builtin_prefetch(Bp + (kt + 2) * BK, 0, 3);
            }
        }

        // compute on current slab: 2 K-steps of 4, 8 WMMAs each
        #pragma unroll
        for (int kk = 0; kk < BK; kk += 4) {
            const int kp = kk + kh * 2;     // this lane's K pair (even -> 8B aligned b64)
            v2f af[4];
            v2f bf[2];
            #pragma unroll
            for (int i = 0; i < 4; ++i)
                af[i] = *(const v2f*)&As[cur][wm + i * 16 + la][kp];
            #pragma unroll
            for (int j = 0; j < 2; ++j)
                bf[j] = *(const v2f*)&Bs[cur][wn + j * 16 + la][kp];
            #pragma unroll
            for (int i = 0; i < 4; ++i)
                #pragma unroll
                for (int j = 0; j < 2; ++j)
                    acc[i][j] = __builtin_amdgcn_wmma_f32_16x16x4_f32(
                        /*neg_a=*/false, af[i],
                        /*neg_b=*/false, bf[j],
                        /*c_mod=*/(short)0, acc[i][j],
                        /*reuse_a=*/false, /*reuse_b=*/false);
        }
    }

    // epilogue: C/D layout — VGPR r holds row (r + 8*kh), lane&15 is the column
    #pragma unroll
    for (int j = 0; j < 2; ++j) {
        const int col = n0 + wn + j * 16 + la;
        const float bv = bias[col];
        #pragma unroll
        for (int i = 0; i < 4; ++i) {
            #pragma unroll
            for (int r = 0; r < 8; ++r) {
                const int row = m0 + wm + i * 16 + kh * 8 + r;
                out[(size_t)row * N_DIM + col] = acc[i][j][r] + bv;
            }
        }
    }
}

extern "C" void kernel_launch(void* const* d_in, const int* in_sizes, int n_in,
                              void* d_out, int out_size, void* d_ws, size_t ws_size,
                              hipStream_t stream) {
    (void)in_sizes; (void)n_in; (void)d_ws; (void)ws_size; (void)out_size;
    const float* x = (const float*)d_in[0];   // [8192, 1024]
    const float* w = (const float*)d_in[1];   // [4096, 1024]
    const float* b = (const float*)d_in[2];   // [4096]
    float* out     = (float*)d_out;           // [8192, 4096]

    dim3 grid(N_DIM / TN, M_DIM / TM);        // (32, 64)
    qlinear_wmma_f32<<<grid, 256, 0, stream>>>(x, w, b, out);
}